// gcn_block_11630771438271
// MI455X (gfx1250) — compile-verified
//
#include <hip/hip_runtime.h>

// GCNConv + ReLU for MI455X (gfx1250, wave32).
// x:[N,128] f32, edge_index:[2,E] int, W:[128,128] f32, b:[128] f32
// out = relu( D^-1/2 (A+I) D^-1/2 (x W) + b )

#define D_DIM 128

typedef float v2f __attribute__((ext_vector_type(2)));
typedef float v8f __attribute__((ext_vector_type(8)));

// ---------------------------------------------------------------- init
__global__ void gcn_init_kernel(float* __restrict__ out, float* __restrict__ deg,
                                int total, int N) {
  int i = blockIdx.x * blockDim.x + threadIdx.x;
  if (i < total) out[i] = 0.0f;
  if (i < N) deg[i] = 1.0f;  // self-loop contributes 1 to each target degree
}

// ---------------------------------------------------------------- degree
__global__ void gcn_degree_kernel(const int* __restrict__ col,
                                  float* __restrict__ deg, int nE) {
  int e = blockIdx.x * blockDim.x + threadIdx.x;
  if (e >= nE) return;
  __hip_atomic_fetch_add(&deg[col[e]], 1.0f, __ATOMIC_RELAXED,
                         __HIP_MEMORY_SCOPE_AGENT);
}

// ---------------------------------------------------------------- deg -> 1/sqrt(deg)
__global__ void gcn_rsqrt_kernel(float* __restrict__ deg, int N) {
  int i = blockIdx.x * blockDim.x + threadIdx.x;
  if (i >= N) return;
  float d = deg[i];
  deg[i] = (d > 0.0f) ? rsqrtf(d) : 0.0f;
}

// ---------------------------------------------------------------- xl = x @ W (fp32 WMMA)
// One wave32 per 16-row strip; 8 column tiles of 16 -> full 16x128 strip.
// A 16x4 f32 fragment: lane m=lane&15 is row M; lane-half picks K pair {0,1}/{2,3}.
// B 4x16 f32 fragment: lane n=lane&15 is col N; lane-half picks same K pair.
// C/D 16x16 f32: VGPR r holds row r (lanes 0-15) / row r+8 (lanes 16-31).
__global__ __launch_bounds__(256) void gcn_gemm_wmma_kernel(
    const float* __restrict__ X, const float* __restrict__ W,
    float* __restrict__ XL, int N) {
  const int lane  = threadIdx.x & 31;
  const int wave  = threadIdx.x >> 5;
  const int tile  = blockIdx.x * 8 + wave;  // 16-row strip index (wave-uniform)
  if (tile * 16 >= N) return;               // whole wave exits together (EXEC all-1 for WMMA)
  const int row0  = tile * 16;
  const int m     = lane & 15;              // row (A) / col (B) within tile
  const int kh    = (lane >> 4) * 2;        // K sub-offset per lane half: 0 or 2

  const float* __restrict__ xrow = X + (size_t)(row0 + m) * D_DIM;

  v8f acc[8] = {};  // 8 col tiles x 8 VGPRs = 64 accumulator VGPRs

  for (int k = 0; k < D_DIM; k += 4) {
    // A fragment: two consecutive K values (8B aligned: k%4==0, kh in {0,2})
    v2f a = *(const v2f*)(xrow + k + kh);
#pragma unroll
    for (int t = 0; t < 8; ++t) {
      const int n = t * 16 + m;
      v2f b;
      b.x = W[(size_t)(k + kh) * D_DIM + n];
      b.y = W[(size_t)(k + kh + 1) * D_DIM + n];
      // 8 args: (neg_a, A, neg_b, B, c_mod, C, reuse_a, reuse_b)
      acc[t] = __builtin_amdgcn_wmma_f32_16x16x4_f32(
          false, a, false, b, (short)0, acc[t], false, false);
    }
  }

  // Store D: VGPR r -> row r + 8*(lane half), col = t*16 + (lane&15)
  const int rbase = (lane >> 4) * 8;
#pragma unroll
  for (int t = 0; t < 8; ++t) {
#pragma unroll
    for (int r = 0; r < 8; ++r) {
      XL[(size_t)(row0 + rbase + r) * D_DIM + t * 16 + m] = acc[t][r];
    }
  }
}

// ---------------------------------------------------------------- edge scatter
// One wave32 per edge: gather xl[row] (hot in 192MB L2), scale by
// dinv[row]*dinv[col], fp32 atomic-add into out[col] (resolves at L2).
__global__ __launch_bounds__(256) void gcn_scatter_kernel(
    const int* __restrict__ ei, const float* __restrict__ xl,
    const float* __restrict__ dinv, float* __restrict__ out, int nE) {
  int gid  = blockIdx.x * blockDim.x + threadIdx.x;
  int e    = gid >> 5;
  int lane = threadIdx.x & 31;
  if (e >= nE) return;
  int r = ei[e];        // source node
  int c = ei[nE + e];   // target node
  float norm = dinv[r] * dinv[c];
  const float4 v = *(const float4*)(xl + (size_t)r * D_DIM + lane * 4);
  float* o = out + (size_t)c * D_DIM + lane * 4;
  __hip_atomic_fetch_add(o + 0, v.x * norm, __ATOMIC_RELAXED, __HIP_MEMORY_SCOPE_AGENT);
  __hip_atomic_fetch_add(o + 1, v.y * norm, __ATOMIC_RELAXED, __HIP_MEMORY_SCOPE_AGENT);
  __hip_atomic_fetch_add(o + 2, v.z * norm, __ATOMIC_RELAXED, __HIP_MEMORY_SCOPE_AGENT);
  __hip_atomic_fetch_add(o + 3, v.w * norm, __ATOMIC_RELAXED, __HIP_MEMORY_SCOPE_AGENT);
}

// ---------------------------------------------------------------- self-loop + bias + relu
__global__ void gcn_finalize_kernel(const float* __restrict__ xl,
                                    const float* __restrict__ dinv,
                                    const float* __restrict__ bias,
                                    float* __restrict__ out, int total) {
  int i = blockIdx.x * blockDim.x + threadIdx.x;
  if (i >= total) return;
  int node = i >> 7;   // / 128
  int d    = i & (D_DIM - 1);
  float di = dinv[node];
  float v  = out[i] + xl[i] * di * di + bias[d];
  out[i]   = fmaxf(v, 0.0f);
}

// ---------------------------------------------------------------- launch
extern "C" void kernel_launch(void* const* d_in, const int* in_sizes, int n_in,
                              void* d_out, int out_size, void* d_ws, size_t ws_size,
                              hipStream_t stream) {
  const float* x  = (const float*)d_in[0];
  const int*   ei = (const int*)d_in[1];
  const float* W  = (const float*)d_in[2];
  const float* b  = (const float*)d_in[3];
  float* out = (float*)d_out;

  const int N  = in_sizes[0] / D_DIM;   // 100000
  const int nE = in_sizes[1] / 2;       // 3200000
  const int total = N * D_DIM;

  // workspace: xl [N*128 f32] then deg/dinv [N f32]
  float* xl  = (float*)d_ws;
  float* deg = xl + (size_t)N * D_DIM;

  const int B256 = 256;

  // 1) zero accumulator (d_out) + set self-loop degree 1
  gcn_init_kernel<<<(total + B256 - 1) / B256, B256, 0, stream>>>(out, deg, total, N);

  // 2) degree over sources (edge_index row 0)
  gcn_degree_kernel<<<(nE + B256 - 1) / B256, B256, 0, stream>>>(ei, deg, nE);

  // 3) deg -> dinv
  gcn_rsqrt_kernel<<<(N + B256 - 1) / B256, B256, 0, stream>>>(deg, N);

  // 4) xl = x @ W  (fp32 WMMA; 8 waves/block, one 16x128 strip per wave)
  const int rowTiles = (N + 15) / 16;
  gcn_gemm_wmma_kernel<<<(rowTiles + 7) / 8, B256, 0, stream>>>(x, W, xl, N);

  // 5) edge gather/scale/scatter-add (one wave per edge)
  long long scatterThreads = (long long)nE * 32;
  gcn_scatter_kernel<<<(unsigned)((scatterThreads + B256 - 1) / B256), B256, 0, stream>>>(
      ei, xl, deg, out, nE);

  // 6) self-loop term + bias + ReLU
  gcn_finalize_kernel<<<(total + B256 - 1) / B256, B256, 0, stream>>>(xl, deg, b, out, total);
}